// ELKAAttention_34308198760602
// MI455X (gfx1250) — compile-verified
//
#include <hip/hip_runtime.h>
#include <hip/hip_bf16.h>
#include <math.h>

typedef __attribute__((ext_vector_type(2))) float v2f;
typedef __attribute__((ext_vector_type(4))) float v4f;
typedef __attribute__((ext_vector_type(8))) float v8f;

#define BATCH 256
#define AREG  196
#define APLUS 197
#define DIM   1024

// CDNA5 has a hardware V_TANH_F32 (TRANS32 pipe). Single instruction,
// branch-free, co-executes with VALU. Fallback: clamped exp-based tanh
// (branch-free, ~8 VALU) — never the divergent libm expansion.
__device__ __forceinline__ float fast_tanhf(float x) {
#if __has_builtin(__builtin_amdgcn_tanh_f32)
    return __builtin_amdgcn_tanh_f32(x);
#elif __has_builtin(__builtin_amdgcn_tanhf)
    return __builtin_amdgcn_tanhf(x);
#else
    float c = fminf(fmaxf(x, -15.f), 15.f);
    float e = __expf(2.f * c);
    return (e - 1.f) / (e + 1.f);
#endif
}

struct GemmArgs {
    const float* X;   // [256, 1024] row-major
    const float* W;   // [1024, 1024] row-major [in, out]
    const float* b;   // [1024]
    float*       Y;   // [256, 1024]
    int          act; // 0 = none, 1 = relu, 2 = tanh
};

// ---------------------------------------------------------------------------
// Full-precision fp32 GEMM on V_WMMA_F32_16X16X4_F32.
// One wave owns a 16(M) x 32(N) output strip (2 accumulators), K-loop step 4.
// Layouts per CDNA5 ISA 7.12.2:
//   A 16x4 : lane L holds A[m = L&15][k0 + 2*(L>>4) + {0,1}]  (v2f)
//   B 4x16 : lane L holds B[k0 + 2*(L>>4) + {0,1}][n = L&15]  (v2f)
//   C 16x16: vgpr v, lane L -> C[m0 + v + 8*(L>>4)][n0 + (L&15)]
// blockIdx.y selects between two independent GEMMs packed in one launch.
// ---------------------------------------------------------------------------
__global__ __launch_bounds__(256) void gemm16x16_pair(GemmArgs ga, GemmArgs gb) {
    GemmArgs g = (blockIdx.y == 0) ? ga : gb;

    const int lane  = threadIdx.x & 31;
    const int wave  = threadIdx.x >> 5;
    const int gw    = blockIdx.x * 8 + wave;   // 0..511
    const int mt    = gw & 15;                 // 16 tiles along M (256/16)
    const int ng    = gw >> 4;                 // 32 groups along N (1024/32)
    const int m0    = mt * 16;
    const int n0    = ng * 32;

    const int arow  = m0 + (lane & 15);
    const int khalf = (lane >> 4) * 2;         // 0 or 2
    const int ncol  = lane & 15;

    const float* Xrow  = g.X + (size_t)arow * DIM;
    const float* Wc0   = g.W + (n0 + ncol);
    const float* Wc1   = g.W + (n0 + 16 + ncol);

    v8f acc0 = {};
    v8f acc1 = {};

#pragma unroll 2
    for (int k = 0; k < DIM; k += 4) {
        const int kk = k + khalf;
        v2f a = *(const v2f*)(Xrow + kk);      // 8B aligned (kk even)
        v2f b0, b1;
        b0.x = Wc0[(size_t)(kk + 0) * DIM];
        b0.y = Wc0[(size_t)(kk + 1) * DIM];
        b1.x = Wc1[(size_t)(kk + 0) * DIM];
        b1.y = Wc1[(size_t)(kk + 1) * DIM];
        // 8 args: (neg_a, A, neg_b, B, c_mod, C, reuse_a, reuse_b)
        acc0 = __builtin_amdgcn_wmma_f32_16x16x4_f32(false, a, false, b0,
                                                     (short)0, acc0, false, false);
        acc1 = __builtin_amdgcn_wmma_f32_16x16x4_f32(false, a, false, b1,
                                                     (short)0, acc1, false, false);
    }

    const float bv0 = g.b[n0 + ncol];
    const float bv1 = g.b[n0 + 16 + ncol];
    const int   rb  = m0 + 8 * (lane >> 4);

#pragma unroll
    for (int v = 0; v < 8; ++v) {
        float y0 = acc0[v] + bv0;
        float y1 = acc1[v] + bv1;
        if (g.act == 1) { y0 = fmaxf(y0, 0.f); y1 = fmaxf(y1, 0.f); }
        else if (g.act == 2) { y0 = fast_tanhf(y0); y1 = fast_tanhf(y1); }
        const size_t r = (size_t)(rb + v) * DIM;
        g.Y[r + n0 + ncol]      = y0;
        g.Y[r + n0 + 16 + ncol] = y1;
    }
}

// ---------------------------------------------------------------------------
// scores[b,a] = sum_d tanh(x[d] + hoe[b,d]) * W_alpha[d] + b_alpha
// x row = fre[b,:] for a==0, else conv_feat_embed[b,a-1,:].
// One wave per (b,a); float4 coalesced loads; wave32 shuffle reduction.
// 51.6M tanh evaluations -> must be single-instruction V_TANH_F32 to stay
// under the 205 MB HBM stream time.
// ---------------------------------------------------------------------------
__global__ __launch_bounds__(256) void scores_kernel(
    const float* __restrict__ cfe, const float* __restrict__ fre,
    const float* __restrict__ hoe, const float* __restrict__ Wa,
    const float* __restrict__ ba,  float* __restrict__ scores) {
    const int task = blockIdx.x * 8 + (threadIdx.x >> 5);
    if (task >= BATCH * APLUS) return;
    const int b    = task / APLUS;
    const int a    = task - b * APLUS;
    const int lane = threadIdx.x & 31;

    const float* xr = (a == 0) ? (fre + (size_t)b * DIM)
                               : (cfe + ((size_t)b * AREG + (a - 1)) * DIM);
    const float* hr = hoe + (size_t)b * DIM;

    float s = 0.f;
#pragma unroll
    for (int i = 0; i < DIM; i += 128) {
        const int d = i + lane * 4;
        v4f x = *(const v4f*)(xr + d);
        v4f h = *(const v4f*)(hr + d);
        v4f w = *(const v4f*)(Wa + d);
        s = fmaf(fast_tanhf(x.x + h.x), w.x, s);
        s = fmaf(fast_tanhf(x.y + h.y), w.y, s);
        s = fmaf(fast_tanhf(x.z + h.z), w.z, s);
        s = fmaf(fast_tanhf(x.w + h.w), w.w, s);
    }
#pragma unroll
    for (int off = 16; off > 0; off >>= 1) s += __shfl_xor(s, off, 32);
    if (lane == 0) scores[task] = s + ba[0];
}

// ---------------------------------------------------------------------------
// Row softmax over 197 entries; one wave per batch row, all in registers.
// ---------------------------------------------------------------------------
__global__ __launch_bounds__(256) void softmax_kernel(
    const float* __restrict__ scores, float* __restrict__ PI) {
    const int b    = blockIdx.x * 8 + (threadIdx.x >> 5);
    const int lane = threadIdx.x & 31;

    float v[7];
    float m = -INFINITY;
#pragma unroll
    for (int i = 0; i < 7; ++i) {
        const int a = lane + i * 32;
        v[i] = (a < APLUS) ? scores[(size_t)b * APLUS + a] : -INFINITY;
        m = fmaxf(m, v[i]);
    }
#pragma unroll
    for (int off = 16; off > 0; off >>= 1) m = fmaxf(m, __shfl_xor(m, off, 32));

    float s = 0.f;
#pragma unroll
    for (int i = 0; i < 7; ++i) { v[i] = __expf(v[i] - m); s += v[i]; }
#pragma unroll
    for (int off = 16; off > 0; off >>= 1) s += __shfl_xor(s, off, 32);
    const float inv = 1.f / s;

#pragma unroll
    for (int i = 0; i < 7; ++i) {
        const int a = lane + i * 32;
        if (a < APLUS) PI[(size_t)b * APLUS + a] = v[i] * inv;
    }
}

// ---------------------------------------------------------------------------
// atten[b,d] = PI[b,0]*fr[b,d] + sum_a PI[b,a+1]*conv_feat[b,a,d] + hol[b,d]
// One block per batch row; PI row staged in LDS; float4 streaming + prefetch.
// ---------------------------------------------------------------------------
__global__ __launch_bounds__(256) void visatt_kernel(
    const float* __restrict__ conv_feat, const float* __restrict__ fr,
    const float* __restrict__ hol, const float* __restrict__ PI,
    float* __restrict__ atten) {
    __shared__ float pis[APLUS];
    const int b = blockIdx.x;
    const int t = threadIdx.x;
    if (t < APLUS) pis[t] = PI[(size_t)b * APLUS + t];
    __syncthreads();

    const int d = t * 4;
    v4f acc;
    {
        v4f f = *(const v4f*)(fr + (size_t)b * DIM + d);
        acc = f * pis[0];
    }
    const float* base = conv_feat + (size_t)b * AREG * DIM + d;
#pragma unroll 4
    for (int a = 0; a < AREG; ++a) {
        const int ap = (a + 8 < AREG) ? a + 8 : AREG - 1;
        __builtin_prefetch(base + (size_t)ap * DIM, 0, 0);
        v4f c = *(const v4f*)(base + (size_t)a * DIM);
        acc += c * pis[a + 1];
    }
    v4f h = *(const v4f*)(hol + (size_t)b * DIM + d);
    acc += h;
    *(v4f*)(atten + (size_t)b * DIM + d) = acc;
}

// ---------------------------------------------------------------------------
extern "C" void kernel_launch(void* const* d_in, const int* in_sizes, int n_in,
                              void* d_out, int out_size, void* d_ws, size_t ws_size,
                              hipStream_t stream) {
    const float* h_out     = (const float*)d_in[0];
    const float* fake_reg  = (const float*)d_in[1];
    const float* conv_feat = (const float*)d_in[2];
    const float* conv_fe   = (const float*)d_in[3];
    const float* W_fr  = (const float*)d_in[4];
    const float* b_fr  = (const float*)d_in[5];
    const float* W_fre = (const float*)d_in[6];
    const float* b_fre = (const float*)d_in[7];
    const float* W_ho  = (const float*)d_in[8];
    const float* b_ho  = (const float*)d_in[9];
    const float* W_hoe = (const float*)d_in[10];
    const float* b_hoe = (const float*)d_in[11];
    const float* W_al  = (const float*)d_in[12];
    const float* b_al  = (const float*)d_in[13];
    const float* W_a2h = (const float*)d_in[14];
    const float* b_a2h = (const float*)d_in[15];
    float* out = (float*)d_out;

    float* ws     = (float*)d_ws;
    const size_t M = (size_t)BATCH * DIM;    // 262144 floats
    float* fr     = ws;
    float* hol    = ws + 1 * M;
    float* fre    = ws + 2 * M;
    float* hoe    = ws + 3 * M;
    float* atten  = ws + 4 * M;
    float* scores = ws + 5 * M;              // 256*197
    float* PI     = scores + (size_t)BATCH * APLUS;

    // Stage 1: fr = relu(fake_region @ W_fr + b)  ||  hol = tanh(h_out @ W_ho + b)
    GemmArgs g0{fake_reg, W_fr, b_fr, fr, 1};
    GemmArgs g1{h_out,    W_ho, b_ho, hol, 2};
    gemm16x16_pair<<<dim3(64, 2), 256, 0, stream>>>(g0, g1);

    // Stage 2: fre = fr @ W_fre + b  ||  hoe = hol @ W_hoe + b
    GemmArgs g2{fr,  W_fre, b_fre, fre, 0};
    GemmArgs g3{hol, W_hoe, b_hoe, hoe, 0};
    gemm16x16_pair<<<dim3(64, 2), 256, 0, stream>>>(g2, g3);

    // Stage 3: attention scores over conv_feat_embed (205 MB stream)
    scores_kernel<<<(BATCH * APLUS + 7) / 8, 256, 0, stream>>>(
        conv_fe, fre, hoe, W_al, b_al, scores);

    // Stage 4: softmax rows
    softmax_kernel<<<BATCH / 8, 256, 0, stream>>>(scores, PI);

    // Stage 5: weighted sum over conv_feat (205 MB stream) + hol
    visatt_kernel<<<BATCH, 256, 0, stream>>>(conv_feat, fr, hol, PI, atten);

    // Stage 6: h = tanh(atten @ W_a2h + b)
    GemmArgs g4{atten, W_a2h, b_a2h, out, 2};
    gemm16x16_pair<<<dim3(64, 1), 256, 0, stream>>>(g4, g4);
}